// DecoderModel_32083405701738
// MI455X (gfx1250) — compile-verified
//
#include <hip/hip_runtime.h>
#include <hip/hip_bf16.h>
#include <math.h>

#define T_IN   128
#define BATCH  64
#define ENC    2048
#define HID    1024
#define NGATES 4096
#define EMBD   128
#define AH     256
#define BOT    128
#define VOCAB  32000
#define T_OUT  32
#define XDIM   (ENC + EMBD)   // 2176

typedef float v2f __attribute__((ext_vector_type(2)));
typedef float v8f __attribute__((ext_vector_type(8)));

__device__ __forceinline__ float sigmoidf_(float x) { return 1.0f / (1.0f + expf(-x)); }

// ---------------------------------------------------------------------------
// Collapse the linear align-MLP to a single vector v[3072] + scalar beff,
// init h = 0 and pred = eos.  v = aW4 @ aW3 @ aW2 @ aW1 (row-vector chain).
// ---------------------------------------------------------------------------
__global__ void prep_kernel(const float* __restrict__ aW1, const float* __restrict__ ab1,
                            const float* __restrict__ aW2, const float* __restrict__ ab2,
                            const float* __restrict__ aW3, const float* __restrict__ ab3,
                            const float* __restrict__ aW4, const float* __restrict__ ab4,
                            const int* __restrict__ eos_tok,
                            float* __restrict__ v, float* __restrict__ beff,
                            float* __restrict__ h, int* __restrict__ pred) {
  __shared__ float u3[AH];
  __shared__ float u2[AH];
  int tid = threadIdx.x;           // 256 threads
  {
    float s = 0.f;
    for (int k = 0; k < AH; ++k) s += aW4[k] * aW3[k * AH + tid];
    u3[tid] = s;
  }
  __syncthreads();
  {
    float s = 0.f;
    for (int k = 0; k < AH; ++k) s += u3[k] * aW2[k * AH + tid];
    u2[tid] = s;
  }
  __syncthreads();
  for (int i = tid; i < 3 * HID; i += AH) {
    float s = 0.f;
    for (int n = 0; n < AH; ++n) s += u2[n] * aW1[n * (3 * HID) + i];
    v[i] = s;
  }
  if (tid == 0) {
    float c3 = ab4[0];
    for (int k = 0; k < AH; ++k) c3 += aW4[k] * ab3[k];
    float c2 = c3;
    for (int m = 0; m < AH; ++m) c2 += u3[m] * ab2[m];
    float c1 = c2;
    for (int n = 0; n < AH; ++n) c1 += u2[n] * ab1[n];
    beff[0] = c1;
  }
  for (int i = tid; i < BATCH * HID; i += AH) h[i] = 0.f;
  if (tid < BATCH) pred[tid] = eos_tok[0];
}

// ---------------------------------------------------------------------------
// Step-invariant half of the attention score: Sp[t,b] = padded[t,b,:] . v[1024:]
// ---------------------------------------------------------------------------
__global__ void sp_kernel(const float* __restrict__ padded, const float* __restrict__ v,
                          float* __restrict__ Sp) {
  int row = blockIdx.x;            // t*BATCH + b
  int tid = threadIdx.x;           // 256
  const float* p  = padded + (size_t)row * ENC;
  const float* vp = v + HID;
  float s = 0.f;
  for (int d = tid; d < ENC; d += 256) s += p[d] * vp[d];
  __shared__ float red[256];
  red[tid] = s; __syncthreads();
  for (int st = 128; st > 0; st >>= 1) {
    if (tid < st) red[tid] += red[tid + st];
    __syncthreads();
  }
  if (tid == 0) Sp[row] = red[0];
}

// ---------------------------------------------------------------------------
// Per-step attention: scores = (Sp + h.v[:1024] + beff) * mask; softmax over t;
// context = sum_t alpha*padded; x = [context | emb_table[pred]]
// ---------------------------------------------------------------------------
__global__ void attn_kernel(const float* __restrict__ padded, const float* __restrict__ mask,
                            const float* __restrict__ emb_table, const int* __restrict__ pred,
                            const float* __restrict__ h, const float* __restrict__ v,
                            const float* __restrict__ beff, const float* __restrict__ Sp,
                            float* __restrict__ x) {
  int b = blockIdx.x;              // 64 blocks
  int tid = threadIdx.x;           // 256
  __shared__ float red[256];
  __shared__ float alpha[T_IN];
  __shared__ float sum_s;
  const float* hb = h + (size_t)b * HID;
  float s = 0.f;
  for (int d = tid; d < HID; d += 256) s += hb[d] * v[d];
  red[tid] = s; __syncthreads();
  for (int st = 128; st > 0; st >>= 1) {
    if (tid < st) red[tid] += red[tid + st];
    __syncthreads();
  }
  float sh = red[0] + beff[0];
  __syncthreads();
  if (tid < T_IN) alpha[tid] = (Sp[tid * BATCH + b] + sh) * mask[tid * BATCH + b];
  __syncthreads();
  if (tid == 0) {                  // 128-point softmax: serial is negligible
    float m = alpha[0];
    for (int t = 1; t < T_IN; ++t) m = fmaxf(m, alpha[t]);
    float su = 0.f;
    for (int t = 0; t < T_IN; ++t) { float e = expf(alpha[t] - m); alpha[t] = e; su += e; }
    sum_s = su;
  }
  __syncthreads();
  float inv = 1.f / sum_s;
  float* xb = x + (size_t)b * XDIM;
  for (int d = tid; d < ENC; d += 256) {
    float c = 0.f;
    for (int t = 0; t < T_IN; ++t) c += alpha[t] * padded[((size_t)t * BATCH + b) * ENC + d];
    xb[d] = c * inv;
  }
  if (tid < EMBD) xb[ENC + tid] = emb_table[(size_t)pred[b] * EMBD + tid];
}

// ---------------------------------------------------------------------------
// WMMA f32 GEMM: out[64,N] = X[64,K] @ W[N,K]^T + b1 (+ b2).
// One wave per N-tile, covering all four 16-row M-tiles: each B fragment load
// feeds FOUR v_wmma_f32_16x16x4_f32 into independent accumulators, so every
// W element is read from L2 exactly once per step (4x less than tile-per-wave).
// Prefetch streams the W row 512B ahead (global_prefetch_b8).
// ---------------------------------------------------------------------------
__global__ void wmma_matmul_kernel(const float* __restrict__ X, const float* __restrict__ W,
                                   const float* __restrict__ b1, const float* __restrict__ b2,
                                   float* __restrict__ out, int N, int K) {
  int ntile = blockIdx.x * blockDim.y + threadIdx.y;    // wave id == N tile
  int lane = threadIdx.x;
  int lr   = lane & 15;
  int half = lane >> 4;
  const float* wrow = W + (size_t)(ntile * 16 + lr) * K;
  const float* x0   = X + (size_t)lr * K;               // M rows lr, lr+16, lr+32, lr+48
  v8f acc0 = {0.f,0.f,0.f,0.f,0.f,0.f,0.f,0.f};
  v8f acc1 = acc0, acc2 = acc0, acc3 = acc0;
  for (int k = 0; k < K; k += 4) {
    int ka = k + 2 * half;
    if ((k & 31) == 0) __builtin_prefetch(wrow + ka + 128, 0, 3);
    v2f wv; wv.x = wrow[ka];              wv.y = wrow[ka + 1];
    v2f a0; a0.x = x0[ka];                a0.y = x0[ka + 1];
    v2f a1; a1.x = x0[16 * K + ka];       a1.y = x0[16 * K + ka + 1];
    v2f a2; a2.x = x0[32 * K + ka];       a2.y = x0[32 * K + ka + 1];
    v2f a3; a3.x = x0[48 * K + ka];       a3.y = x0[48 * K + ka + 1];
    acc0 = __builtin_amdgcn_wmma_f32_16x16x4_f32(false, a0, false, wv, (short)0, acc0, false, false);
    acc1 = __builtin_amdgcn_wmma_f32_16x16x4_f32(false, a1, false, wv, (short)0, acc1, false, false);
    acc2 = __builtin_amdgcn_wmma_f32_16x16x4_f32(false, a2, false, wv, (short)0, acc2, false, false);
    acc3 = __builtin_amdgcn_wmma_f32_16x16x4_f32(false, a3, false, wv, (short)0, acc3, false, false);
  }
  int n = ntile * 16 + lr;
  float bias = b1[n] + (b2 ? b2[n] : 0.f);
  // C/D layout: VGPR j -> M = j (+8 for upper lane half)
  for (int j = 0; j < 8; ++j) {
    int m = j + 8 * half;
    out[(size_t)(m)      * N + n] = acc0[j] + bias;
    out[(size_t)(m + 16) * N + n] = acc1[j] + bias;
    out[(size_t)(m + 32) * N + n] = acc2[j] + bias;
    out[(size_t)(m + 48) * N + n] = acc3[j] + bias;
  }
}

// ---------------------------------------------------------------------------
// Degenerate LSTM cell (zero state): h = sigmoid(o) * tanh(sigmoid(i)*tanh(g))
// gate order i,f,g,o in chunks of HID; f unused since c_prev == 0.
// ---------------------------------------------------------------------------
__global__ void lstm_act_kernel(const float* __restrict__ gates, float* __restrict__ h) {
  int idx = blockIdx.x * blockDim.x + threadIdx.x;
  if (idx >= BATCH * HID) return;
  int b = idx / HID, j = idx - b * HID;
  const float* g = gates + (size_t)b * NGATES;
  float gi = g[j];
  float gg = g[2 * HID + j];
  float go = g[3 * HID + j];
  float c  = sigmoidf_(gi) * tanhf(gg);
  h[idx] = sigmoidf_(go) * tanhf(c);
}

// ---------------------------------------------------------------------------
// Softmax over VOCAB + argmax (first-max tie-break, matching jnp.argmax).
// ---------------------------------------------------------------------------
__global__ void softmax_argmax_kernel(const float* __restrict__ logits,
                                      float* __restrict__ out, int* __restrict__ pred) {
  int b = blockIdx.x;              // 64 blocks
  int tid = threadIdx.x;           // 256
  __shared__ float rv[256];
  __shared__ int   ri[256];
  const float* row = logits + (size_t)b * VOCAB;
  float bm = -3.402823466e38f; int bi = 0;
  for (int n = tid; n < VOCAB; n += 256) {
    float vl = row[n];
    if (vl > bm) { bm = vl; bi = n; }
  }
  rv[tid] = bm; ri[tid] = bi; __syncthreads();
  for (int st = 128; st > 0; st >>= 1) {
    if (tid < st) {
      if (rv[tid + st] > rv[tid] || (rv[tid + st] == rv[tid] && ri[tid + st] < ri[tid])) {
        rv[tid] = rv[tid + st]; ri[tid] = ri[tid + st];
      }
    }
    __syncthreads();
  }
  float mx = rv[0]; int amax = ri[0];
  __syncthreads();
  float ps = 0.f;
  for (int n = tid; n < VOCAB; n += 256) ps += expf(row[n] - mx);
  rv[tid] = ps; __syncthreads();
  for (int st = 128; st > 0; st >>= 1) {
    if (tid < st) rv[tid] += rv[tid + st];
    __syncthreads();
  }
  float inv = 1.f / rv[0];
  for (int n = tid; n < VOCAB; n += 256) out[(size_t)b * VOCAB + n] = expf(row[n] - mx) * inv;
  if (tid == 0) pred[b] = amax;
}

// ---------------------------------------------------------------------------
extern "C" void kernel_launch(void* const* d_in, const int* in_sizes, int n_in,
                              void* d_out, int out_size, void* d_ws, size_t ws_size,
                              hipStream_t stream) {
  const float* padded = (const float*)d_in[0];
  const float* mask   = (const float*)d_in[1];
  const float* embt   = (const float*)d_in[2];
  const float* aW1 = (const float*)d_in[3];
  const float* ab1 = (const float*)d_in[4];
  const float* aW2 = (const float*)d_in[5];
  const float* ab2 = (const float*)d_in[6];
  const float* aW3 = (const float*)d_in[7];
  const float* ab3 = (const float*)d_in[8];
  const float* aW4 = (const float*)d_in[9];
  const float* ab4 = (const float*)d_in[10];

  const float *Wih[4], *bih[4], *bhh[4];
  const float *Wb, *bb_, *Wl, *bl; const int* eos;
  if (n_in >= 34) {
    // tuples expanded: Wih 11-14, Whh 15-18 (unused: h==0), bih 19-22, bhh 23-26
    for (int l = 0; l < 4; ++l) Wih[l] = (const float*)d_in[11 + l];
    for (int l = 0; l < 4; ++l) bih[l] = (const float*)d_in[19 + l];
    for (int l = 0; l < 4; ++l) bhh[l] = (const float*)d_in[23 + l];
    Wb  = (const float*)d_in[27]; bb_ = (const float*)d_in[28];
    Wl  = (const float*)d_in[29]; bl  = (const float*)d_in[30];
    eos = (const int*)d_in[31];
  } else {
    // tuples flattened+concatenated into one entry each
    const float* wih = (const float*)d_in[11];
    Wih[0] = wih;
    Wih[1] = Wih[0] + (size_t)NGATES * XDIM;
    Wih[2] = Wih[1] + (size_t)NGATES * HID;
    Wih[3] = Wih[2] + (size_t)NGATES * HID;
    const float* bi = (const float*)d_in[13];   // d_in[12] = Whh (unused)
    const float* bh = (const float*)d_in[14];
    for (int l = 0; l < 4; ++l) { bih[l] = bi + (size_t)NGATES * l; bhh[l] = bh + (size_t)NGATES * l; }
    Wb  = (const float*)d_in[15]; bb_ = (const float*)d_in[16];
    Wl  = (const float*)d_in[17]; bl  = (const float*)d_in[18];
    eos = (const int*)d_in[19];
  }

  // workspace layout (floats)
  float* ws     = (float*)d_ws;
  float* v      = ws;                                   // 3072
  float* beff   = v + 3 * HID;                          // 1 (padded to 4)
  float* Sp     = beff + 4;                             // 8192
  float* h      = Sp + T_IN * BATCH;                    // 65536
  float* x      = h + BATCH * HID;                      // 139264
  float* gates  = x + BATCH * XDIM;                     // 262144
  float* bott   = gates + BATCH * NGATES;               // 8192
  float* logits = bott + BATCH * BOT;                   // 2048000
  int*   pred   = (int*)(logits + (size_t)BATCH * VOCAB);

  prep_kernel<<<1, 256, 0, stream>>>(aW1, ab1, aW2, ab2, aW3, ab3, aW4, ab4,
                                     eos, v, beff, h, pred);
  sp_kernel<<<T_IN * BATCH, 256, 0, stream>>>(padded, v, Sp);

  float* out = (float*)d_out;
  for (int s = 0; s < T_OUT; ++s) {
    attn_kernel<<<BATCH, 256, 0, stream>>>(padded, mask, embt, pred, h, v, beff, Sp, x);
    for (int l = 0; l < 4; ++l) {
      int K = (l == 0) ? XDIM : HID;
      const float* in = (l == 0) ? x : h;
      int nwaves = NGATES / 16;                          // 256 waves (one per N tile)
      wmma_matmul_kernel<<<dim3(nwaves / 8), dim3(32, 8), 0, stream>>>(
          in, Wih[l], bih[l], bhh[l], gates, NGATES, K);
      lstm_act_kernel<<<(BATCH * HID) / 256, 256, 0, stream>>>(gates, h);
    }
    {
      int nwaves = BOT / 16;                             // 8 waves
      wmma_matmul_kernel<<<dim3(1), dim3(32, nwaves), 0, stream>>>(
          h, Wb, bb_, nullptr, bott, BOT, HID);
    }
    {
      int nwaves = VOCAB / 16;                           // 2000 waves
      wmma_matmul_kernel<<<dim3(nwaves / 8), dim3(32, 8), 0, stream>>>(
          bott, Wl, bl, nullptr, logits, VOCAB, BOT);
    }
    softmax_argmax_kernel<<<BATCH, 256, 0, stream>>>(
        logits, out + (size_t)s * BATCH * VOCAB, pred);
  }
}